// ModelAtten_38173669327416
// MI455X (gfx1250) — compile-verified
//
#include <hip/hip_runtime.h>
#include <hip/hip_bf16.h>

typedef __attribute__((ext_vector_type(2))) float v2f;
typedef __attribute__((ext_vector_type(4))) float v4f;
typedef __attribute__((ext_vector_type(8))) float v8f;

#define HD 128        // feature dim (F == H == 128)
#define PPART 64      // partitions

// ---------------------------------------------------------------- utilities
__global__ void zero_kernel(float* __restrict__ p, size_t n) {
    size_t i = (size_t)blockIdx.x * blockDim.x + threadIdx.x;
    size_t st = (size_t)gridDim.x * blockDim.x;
    for (; i < n; i += st) p[i] = 0.0f;
}

__global__ void fill1_kernel(float* __restrict__ p, int n) {
    int i = blockIdx.x * blockDim.x + threadIdx.x;
    if (i < n) p[i] = 1.0f;
}

// ---------------------------------------------------------------- degree
__global__ void deg_count_kernel(const int* __restrict__ dst, float* __restrict__ deg, int E) {
    int i = blockIdx.x * blockDim.x + threadIdx.x;
    int st = gridDim.x * blockDim.x;
    for (; i < E; i += st) atomicAdd(&deg[dst[i]], 1.0f);
}

__global__ void deg_inv_kernel(float* __restrict__ deg, int n) {
    int i = blockIdx.x * blockDim.x + threadIdx.x;
    if (i < n) deg[i] = rsqrtf(fmaxf(deg[i], 1.0f));
}

// ---------------------------------------------------------------- WMMA GEMM: Y[N,128] = X[N,128] @ W[128,128]
// block = 256 threads (8 waves); each wave owns a 16-row band x 128 cols
// f32 WMMA 16x16x4, K accumulated 4 at a time (ISA 7.12.2 32-bit layouts)
__global__ __launch_bounds__(256) void gemm_wmma_kernel(const float* __restrict__ X,
                                                        const float* __restrict__ W,
                                                        float* __restrict__ Y, int Nrows) {
    __shared__ float lw[HD * HD];   // 64 KB: whole weight matrix
    for (int i = threadIdx.x; i < HD * HD; i += 256) lw[i] = W[i];
    __syncthreads();

    const int wave  = threadIdx.x >> 5;
    const int lane  = threadIdx.x & 31;
    const int m     = lane & 15;     // A row-in-tile / B,C col-in-tile
    const int khalf = lane >> 4;     // 0 -> K{0,1}, 1 -> K{2,3}
    const int row0  = (blockIdx.x * 8 + wave) * 16;

    // Branchless ragged-tail handling: clamp the row, zero via mask multiply.
    const int   r     = row0 + m;
    const int   rc    = (r < Nrows) ? r : (Nrows - 1);
    const float amask = (r < Nrows) ? 1.0f : 0.0f;
    const float* __restrict__ xrow = X + (size_t)rc * HD + khalf * 2;

    v8f acc[8];
    const v8f vz = {0.f, 0.f, 0.f, 0.f, 0.f, 0.f, 0.f, 0.f};
#pragma unroll
    for (int t = 0; t < 8; ++t) acc[t] = vz;

    for (int k = 0; k < HD; k += 4) {
        // A fragment (16x4 f32): lanes 0-15 hold K=k..k+1, lanes 16-31 hold K=k+2..k+3
        v2f a = *(const v2f*)(xrow + k);   // unconditional, 8B-aligned b64 load
        a.x *= amask;
        a.y *= amask;
#pragma unroll
        for (int t = 0; t < 8; ++t) {
            const int n0 = t * 16;
            // B fragment (4x16 f32), mirrored layout: lane -> col n0+m, khalf -> K pair
            v2f b;
            b.x = lw[(k + khalf * 2 + 0) * HD + n0 + m];
            b.y = lw[(k + khalf * 2 + 1) * HD + n0 + m];
            acc[t] = __builtin_amdgcn_wmma_f32_16x16x4_f32(
                false, a, false, b, (short)0, acc[t], false, false);
        }
    }

    // C layout: VGPR v -> row (v + 8*khalf), col m (within tile)
#pragma unroll
    for (int v = 0; v < 8; ++v) {
        int rr = row0 + khalf * 8 + v;
        if (rr < Nrows) {
#pragma unroll
            for (int t = 0; t < 8; ++t)
                Y[(size_t)rr * HD + t * 16 + m] = acc[t][v];
        }
    }
}

// ---------------------------------------------------------------- hs = h * inv_sqrt[row]
__global__ void scale_rows_kernel(float* __restrict__ h, const float* __restrict__ inv, size_t total) {
    size_t i = (size_t)blockIdx.x * blockDim.x + threadIdx.x;
    size_t st = (size_t)gridDim.x * blockDim.x;
    for (; i < total; i += st) h[i] *= inv[i >> 7];
}

// ---------------------------------------------------------------- edge scatter: acc[d] += hs[s], wave per edge
__global__ __launch_bounds__(256) void edge_scatter_kernel(const int* __restrict__ src,
                                                           const int* __restrict__ dst,
                                                           const float* __restrict__ hs,
                                                           float* __restrict__ acc, int E) {
    const int lane = threadIdx.x & 31;
    int wid = (blockIdx.x * blockDim.x + threadIdx.x) >> 5;
    const int nw = (gridDim.x * blockDim.x) >> 5;
    for (int e = wid; e < E; e += nw) {
        int s = src[e];
        int d = dst[e];
        v4f v = ((const v4f*)(hs + (size_t)s * HD))[lane];
        float* ap = acc + (size_t)d * HD + lane * 4;
        atomicAdd(ap + 0, v.x);
        atomicAdd(ap + 1, v.y);
        atomicAdd(ap + 2, v.z);
        atomicAdd(ap + 3, v.w);
    }
}

// ---------------------------------------------------------------- agg = (acc + hs) * inv_sqrt[row] + bias[col]
__global__ void finalize_kernel(float* __restrict__ acc, const float* __restrict__ hs,
                                const float* __restrict__ inv, const float* __restrict__ bias,
                                size_t total) {
    size_t i = (size_t)blockIdx.x * blockDim.x + threadIdx.x;
    size_t st = (size_t)gridDim.x * blockDim.x;
    for (; i < total; i += st)
        acc[i] = (acc[i] + hs[i]) * inv[i >> 7] + bias[i & (HD - 1)];
}

// ---------------------------------------------------------------- batch norm (column stats over N rows)
__global__ __launch_bounds__(128) void bn_stats_kernel(const float* __restrict__ h,
                                                       float* __restrict__ sum,
                                                       float* __restrict__ sumsq, int Nrows) {
    const int f = threadIdx.x;
    const int r0 = blockIdx.x * 64;
    float s = 0.f, s2 = 0.f;
    for (int i = 0; i < 64; ++i) {
        int r = r0 + i;
        if (r < Nrows) {
            float v = h[(size_t)r * HD + f];
            s += v;
            s2 += v * v;
        }
    }
    atomicAdd(&sum[f], s);
    atomicAdd(&sumsq[f], s2);
}

__global__ void bn_final_kernel(const float* __restrict__ sum, const float* __restrict__ sumsq,
                                float* __restrict__ mu, float* __restrict__ rs, int Nrows) {
    int f = threadIdx.x;
    float inv_n = 1.0f / (float)Nrows;
    float m = sum[f] * inv_n;
    float var = sumsq[f] * inv_n - m * m;
    mu[f] = m;
    rs[f] = rsqrtf(fmaxf(var, 0.f) + 1e-5f);
}

__global__ void bn_apply_kernel(const float* __restrict__ h, const float* __restrict__ mu,
                                const float* __restrict__ rs, float* __restrict__ xout, size_t total) {
    size_t i = (size_t)blockIdx.x * blockDim.x + threadIdx.x;
    size_t st = (size_t)gridDim.x * blockDim.x;
    for (; i < total; i += st) {
        int f = (int)(i & (HD - 1));
        xout[i] = fmaxf((h[i] - mu[f]) * rs[f], 0.f);
    }
}

// ---------------------------------------------------------------- partition pooling
__global__ void pool_kernel(const float* __restrict__ x, const float* __restrict__ cv,
                            const int* __restrict__ pid, float* __restrict__ core,
                            float* __restrict__ ncore, size_t total) {
    size_t i = (size_t)blockIdx.x * blockDim.x + threadIdx.x;
    size_t st = (size_t)gridDim.x * blockDim.x;
    for (; i < total; i += st) {
        int v = (int)(i >> 7);
        int f = (int)(i & (HD - 1));
        float xv = x[i];
        float c = cv[v];
        int p = pid[v];
        atomicAdd(&core[(size_t)p * HD + f], xv * c);
        atomicAdd(&ncore[(size_t)p * HD + f], xv * (1.0f - c));
    }
}

// ---------------------------------------------------------------- final MLP: z=[x_curr|core|ncore] (P,384) -> relu(z@W1+b1) -> @W2+b2
__global__ __launch_bounds__(128) void mlp_kernel(const float* __restrict__ x,
                                                  const int* __restrict__ curr,
                                                  const float* __restrict__ core,
                                                  const float* __restrict__ ncore,
                                                  const float* __restrict__ W1,
                                                  const float* __restrict__ b1,
                                                  const float* __restrict__ W2,
                                                  const float* __restrict__ b2,
                                                  float* __restrict__ out) {
    __shared__ float xc[HD];
    __shared__ float red[HD];
    const int j = threadIdx.x;
    xc[j] = x[(size_t)(*curr) * HD + j];
    __syncthreads();

    for (int p = 0; p < PPART; ++p) {
        float a = b1[j];
        for (int k = 0; k < HD; ++k)          a += xc[k]                   * W1[(size_t)k * HD + j];
        for (int k = 0; k < HD; ++k)          a += core[(size_t)p*HD + k]  * W1[(size_t)(HD + k) * HD + j];
        for (int k = 0; k < HD; ++k)          a += ncore[(size_t)p*HD + k] * W1[(size_t)(2*HD + k) * HD + j];
        float h1 = fmaxf(a, 0.f);
        red[j] = h1 * W2[j];
        __syncthreads();
        for (int s = HD / 2; s > 0; s >>= 1) {
            if (j < s) red[j] += red[j + s];
            __syncthreads();
        }
        if (j == 0) out[p] = red[0] + b2[0];
        __syncthreads();
    }
}

// ---------------------------------------------------------------- launcher
extern "C" void kernel_launch(void* const* d_in, const int* in_sizes, int n_in,
                              void* d_out, int out_size, void* d_ws, size_t ws_size,
                              hipStream_t stream) {
    const float* x_in  = (const float*)d_in[0];
    const float* cv    = (const float*)d_in[1];
    const float* W0    = (const float*)d_in[2];
    const float* b0    = (const float*)d_in[3];
    const float* W1    = (const float*)d_in[4];
    const float* b1    = (const float*)d_in[5];
    const float* lin1W = (const float*)d_in[6];
    const float* lin1b = (const float*)d_in[7];
    const float* lin2W = (const float*)d_in[8];
    const float* lin2b = (const float*)d_in[9];
    const int*   eidx  = (const int*)d_in[10];
    const int*   pid   = (const int*)d_in[11];
    const int*   curr  = (const int*)d_in[12];

    const int Nn = in_sizes[1];          // 50000
    const int E  = in_sizes[10] / 2;     // 640000
    const int* src = eidx;
    const int* dst = eidx + E;
    const size_t nf = (size_t)Nn * HD;

    float* ws     = (float*)d_ws;
    float* bufA   = ws;                  // x (layer input after layer0)
    float* bufB   = bufA + nf;           // h -> hs
    float* bufC   = bufB + nf;           // acc -> agg
    float* inv    = bufC + nf;           // N: deg -> inv_sqrt
    float* bnsum  = inv + Nn;
    float* bnsq   = bnsum + HD;
    float* bnmu   = bnsq + HD;
    float* bnrs   = bnmu + HD;
    float* coreb  = bnrs + HD;           // P x H
    float* ncoreb = coreb + (size_t)PPART * HD;

    const int T = 256;
    const int gN   = (Nn + T - 1) / T;
    const int gNF  = 2048;                         // grid-stride over N*128 elements
    const int gEdg = 4096;                         // wave-per-edge grid-stride
    const int gGem = (Nn + 127) / 128;             // 128 rows / block
    const int gBN  = (Nn + 63) / 64;

    // degrees: deg = 1 (self loop) + in-degree, then inv_sqrt in place
    fill1_kernel<<<gN, T, 0, stream>>>(inv, Nn);
    deg_count_kernel<<<2048, T, 0, stream>>>(dst, inv, E);
    deg_inv_kernel<<<gN, T, 0, stream>>>(inv, Nn);

    const float* xcur = x_in;
    for (int layer = 0; layer < 2; ++layer) {
        const float* W = layer ? W1 : W0;
        const float* b = layer ? b1 : b0;

        gemm_wmma_kernel<<<gGem, 256, 0, stream>>>(xcur, W, bufB, Nn);          // h = x @ W
        scale_rows_kernel<<<gNF, T, 0, stream>>>(bufB, inv, nf);                // hs = h * invs[v]
        zero_kernel<<<gNF, T, 0, stream>>>(bufC, nf);                           // acc = 0
        edge_scatter_kernel<<<gEdg, T, 0, stream>>>(src, dst, bufB, bufC, E);   // acc[d] += hs[s]
        finalize_kernel<<<gNF, T, 0, stream>>>(bufC, bufB, inv, b, nf);         // (acc+hs)*invs + b

        zero_kernel<<<1, 256, 0, stream>>>(bnsum, 2 * HD);                      // sum & sumsq
        bn_stats_kernel<<<gBN, HD, 0, stream>>>(bufC, bnsum, bnsq, Nn);
        bn_final_kernel<<<1, HD, 0, stream>>>(bnsum, bnsq, bnmu, bnrs, Nn);
        bn_apply_kernel<<<gNF, T, 0, stream>>>(bufC, bnmu, bnrs, bufA, nf);     // relu(BN(.))
        xcur = bufA;
    }

    zero_kernel<<<64, 256, 0, stream>>>(coreb, 2 * (size_t)PPART * HD);
    pool_kernel<<<gNF, T, 0, stream>>>(bufA, cv, pid, coreb, ncoreb, nf);
    mlp_kernel<<<1, HD, 0, stream>>>(bufA, curr, coreb, ncoreb,
                                     lin1W, lin1b, lin2W, lin2b, (float*)d_out);
}